// GraphAttentionLayer_72413148611006
// MI455X (gfx1250) — compile-verified
//
#include <hip/hip_runtime.h>
#include <hip/hip_bf16.h>
#include <math.h>

typedef __attribute__((ext_vector_type(2))) float v2f;
typedef __attribute__((ext_vector_type(8))) float v8f;

// Computes Y[h] = X[h] @ W[h]^T (+ bias[h]) with V_WMMA_F32_16X16X4_F32.
// X: per-head base = X + head*xHeadStride, rows of 128 f32 (row-major).
// W: [head][128][128] row-major as W[n][k]  (so B-matrix B[k][n] = W[n][k]).
// Y: [head][M][128].
// One wave (32 threads) computes a 16-row x 128-col output strip.
__global__ __launch_bounds__(32)
void gat_gemm_xwT_kernel(const float* __restrict__ X, size_t xHeadStride,
                         const float* __restrict__ W,
                         const float* __restrict__ bias,
                         float* __restrict__ Y, int M)
{
    const int head = blockIdx.y;
    const int m0   = blockIdx.x * 16;
    const int lane = threadIdx.x & 31;
    const int half = lane >> 4;   // 0 -> K{0,1}, 1 -> K{2,3} within a k-step
    const int lr   = lane & 15;

    const float* Xh = X + (size_t)head * xHeadStride;
    const float* Wh = W + (size_t)head * (128 * 128);
    float*       Yh = Y + (size_t)head * (size_t)M * 128;

    int ar = m0 + lr;                 // A-matrix row owned by this lane
    if (ar > M - 1) ar = M - 1;       // clamp (M=50000 is a multiple of 16; safety only)
    const float* arow = Xh + (size_t)ar * 128;

    v8f acc[8] = {};                  // 8 n-tiles of 16 -> full 128-wide strip

    for (int k = 0; k < 128; k += 4) {
        // A frag: lane holds A[m=lr][k+2*half .. k+2*half+1]  (8B-aligned)
        v2f a = *(const v2f*)(arow + k + 2 * half);
#pragma unroll
        for (int t = 0; t < 8; ++t) {
            // B frag: B[kk][n] = W[n][kk]; n = t*16 + lr -> same row-major pattern
            const float* brow = Wh + (size_t)(t * 16 + lr) * 128;
            v2f b = *(const v2f*)(brow + k + 2 * half);
            acc[t] = __builtin_amdgcn_wmma_f32_16x16x4_f32(
                false, a, false, b, (short)0, acc[t], false, false);
        }
    }

    // D layout: VGPR r -> rows (r) [lanes 0-15] and (r+8) [lanes 16-31], col = lr
#pragma unroll
    for (int t = 0; t < 8; ++t) {
        const int n = t * 16 + lr;
        const float bv = bias ? bias[(size_t)head * 128 + n] : 0.0f;
#pragma unroll
        for (int r = 0; r < 8; ++r) {
            const int m = m0 + r + 8 * half;
            if (m < M) Yh[(size_t)m * 128 + n] = acc[t][r] + bv;
        }
    }
}

// One wave per edge: score = sum_h dot(H[h][row], G[h][col]); out = sigmoid(score/2).
// Lane l loads floats [4l..4l+3] of each 128-f32 vector (fully coalesced 512B/vector).
// Gathers hit L2: H+G = 102.4 MB < 192 MB global L2.
__global__ __launch_bounds__(256)
void gat_edge_kernel(const int* __restrict__ rows, const int* __restrict__ cols,
                     const float* __restrict__ H, const float* __restrict__ G,
                     float* __restrict__ out, int nEdges, int M)
{
    const int lane          = threadIdx.x & 31;
    const int wavesPerBlock = blockDim.x >> 5;
    const int w0            = blockIdx.x * wavesPerBlock + (threadIdx.x >> 5);
    const int nW            = gridDim.x * wavesPerBlock;
    const size_t headStride = (size_t)M * 128;

    for (int e = w0; e < nEdges; e += nW) {
        const int r = rows[e];
        const int c = cols[e];
        float s = 0.0f;
#pragma unroll
        for (int h = 0; h < 2; ++h) {
            const float4 a = ((const float4*)(H + h * headStride + (size_t)r * 128))[lane];
            const float4 b = ((const float4*)(G + h * headStride + (size_t)c * 128))[lane];
            s = fmaf(a.x, b.x, s);
            s = fmaf(a.y, b.y, s);
            s = fmaf(a.z, b.z, s);
            s = fmaf(a.w, b.w, s);
        }
#pragma unroll
        for (int off = 16; off; off >>= 1)
            s += __shfl_xor(s, off, 32);
        if (lane == 0)
            out[e] = 1.0f / (1.0f + expf(-0.5f * s));
    }
}

extern "C" void kernel_launch(void* const* d_in, const int* in_sizes, int n_in,
                              void* d_out, int out_size, void* d_ws, size_t ws_size,
                              hipStream_t stream)
{
    (void)n_in; (void)out_size; (void)ws_size;

    const float* x     = (const float*)d_in[0];   // (M, 128) f32
    const int*   ei    = (const int*)  d_in[1];   // (2, E) indices
    const float* W_lin = (const float*)d_in[2];   // (2, 128, 128) f32
    const float* b_lin = (const float*)d_in[3];   // (2, 128) f32
    const float* W_att = (const float*)d_in[4];   // (2, 128, 128) f32
    float*       out   = (float*)d_out;           // (E,) f32

    const int M      = in_sizes[0] / 128;         // 50000
    const int nEdges = in_sizes[1] / 2;           // 400000

    // Workspace: H (2*M*128 f32 = 51.2 MB) then G (51.2 MB); fully overwritten each call.
    float* H = (float*)d_ws;
    float* G = H + (size_t)2 * (size_t)M * 128;

    dim3 gridG((M + 15) / 16, 2);
    // H[h] = x @ W_lin[h]^T + b_lin[h]   (x shared across heads -> head stride 0)
    gat_gemm_xwT_kernel<<<gridG, 32, 0, stream>>>(x, (size_t)0, W_lin, b_lin, H, M);
    // G[h] = H[h] @ W_att[h]^T
    gat_gemm_xwT_kernel<<<gridG, 32, 0, stream>>>(H, (size_t)M * 128, W_att, nullptr, G, M);

    const int blocks = (nEdges + 7) / 8;          // 8 waves/block, 1 edge/wave
    gat_edge_kernel<<<blocks, 256, 0, stream>>>(ei, ei + nEdges, H, G, out, nEdges, M);
}